// ProductionEMDDenoisedInSARModel_85779086835975
// MI455X (gfx1250) — compile-verified
//
#include <hip/hip_runtime.h>

typedef __attribute__((ext_vector_type(2))) float v2f;
typedef __attribute__((ext_vector_type(8))) float v8f;

#define SMOOTH_S 0.12f
#define BS_COLS  1024   // padded time columns (>= N_TIME, multiple of 16)
#define TWO_PI_F 6.283185307179586f

// ---------------------------------------------------------------------------
// Kernel 1: per-station coefficients (spatial smoothing folded into A·cosφ /
// A·sinφ pairs; no atan2 needed).  coef[s][16], rows padded with zeros.
// ---------------------------------------------------------------------------
__global__ void coef_kernel(const float* __restrict__ c0,
                            const float* __restrict__ trend,
                            const float* __restrict__ samp,
                            const float* __restrict__ sphase,
                            const float* __restrict__ ramp,
                            const float* __restrict__ rphase,
                            const int*   __restrict__ nbidx,
                            const float* __restrict__ nbw,
                            float* __restrict__ coef,
                            int nstations, int npad)
{
    int s = blockIdx.x * blockDim.x + threadIdx.x;
    if (s >= npad) return;

    float out[16];
    #pragma unroll
    for (int q = 0; q < 16; ++q) out[q] = 0.0f;

    if (s < nstations) {
        int   nb[8];
        float w[8];
        #pragma unroll
        for (int j = 0; j < 8; ++j) { nb[j] = nbidx[s * 8 + j]; w[j] = nbw[s * 8 + j]; }

        out[0] = c0[s];
        out[1] = trend[s];

        // 3 seasonal oscillators: smooth amplitude linearly, phase circularly.
        #pragma unroll
        for (int i = 0; i < 3; ++i) {
            float a  = samp[s * 3 + i];
            float p  = sphase[s * 3 + i];
            float sa = (1.0f - SMOOTH_S) * a;
            float re = (1.0f - SMOOTH_S) * cosf(p);
            float im = (1.0f - SMOOTH_S) * sinf(p);
            #pragma unroll
            for (int j = 0; j < 8; ++j) {
                float sw = SMOOTH_S * w[j];
                float an = samp[nb[j] * 3 + i];
                float pn = sphase[nb[j] * 3 + i];
                sa = fmaf(sw, an, sa);
                re = fmaf(sw, cosf(pn), re);
                im = fmaf(sw, sinf(pn), im);
            }
            // cos(atan2(im,re)) = re/h ; sin(atan2(im,re)) = im/h
            float h    = sqrtf(fmaf(re, re, im * im));
            float invh = (h > 0.0f) ? (1.0f / h) : 0.0f;
            out[2 + 2 * i] = sa * re * invh;  // multiplies sin(w t)
            out[3 + 2 * i] = sa * im * invh;  // multiplies cos(w t)
        }
        // 2 residual oscillators (no smoothing).
        #pragma unroll
        for (int i = 0; i < 2; ++i) {
            float a = ramp[s * 2 + i];
            float p = rphase[s * 2 + i];
            out[8 + 2 * i] = a * cosf(p);
            out[9 + 2 * i] = a * sinf(p);
        }
        // out[12..15] stay 0 (K padding to 16)
    }

    float4* dst = (float4*)(coef + (size_t)s * 16);
    #pragma unroll
    for (int q = 0; q < 4; ++q) {
        dst[q] = make_float4(out[4 * q + 0], out[4 * q + 1],
                             out[4 * q + 2], out[4 * q + 3]);
    }
}

// ---------------------------------------------------------------------------
// Kernel 2: basis functions per time column, stored as (k-pair, col, 2) so a
// lane fetches its two K values with one b64 load per WMMA step.
//   pair 0 = (1, t); pair 1+f = (sin(2πt/P_f), cos(2πt/P_f)); pairs 6,7 = 0.
// ---------------------------------------------------------------------------
__global__ void basis_kernel(const float* __restrict__ tv,
                             const float* __restrict__ periods,
                             const float* __restrict__ rper,
                             float* __restrict__ bs,
                             int ntime)
{
    int i = blockIdx.x * blockDim.x + threadIdx.x;
    if (i >= BS_COLS) return;
    bool  ok = i < ntime;
    float t  = ok ? tv[i] : 0.0f;

    float P[5] = { periods[0], periods[1], periods[2], rper[0], rper[1] };

    float2* out2 = (float2*)bs;
    out2[0 * BS_COLS + i] = make_float2(ok ? 1.0f : 0.0f, t);
    #pragma unroll
    for (int f = 0; f < 5; ++f) {
        float ang = TWO_PI_F * t / P[f];
        float sv  = ok ? sinf(ang) : 0.0f;
        float cv  = ok ? cosf(ang) : 0.0f;
        out2[(1 + f) * BS_COLS + i] = make_float2(sv, cv);
    }
    out2[6 * BS_COLS + i] = make_float2(0.0f, 0.0f);
    out2[7 * BS_COLS + i] = make_float2(0.0f, 0.0f);
}

// ---------------------------------------------------------------------------
// Kernel 3: the GEMM.  One wave per 16x16 output tile, 4x
// V_WMMA_F32_16X16X4_F32 over K=16.  Non-temporal stores for the 400MB
// streaming output (store-BW bound: ~17us at 23.3 TB/s).  The common
// M-aligned case (nstations % 16 == 0) takes a scalar-branch fast path with
// an unguarded run of 8 NT stores (single v_cmpx on col only).
// ---------------------------------------------------------------------------
__global__ void __launch_bounds__(256)
wmma_gemm_kernel(const float* __restrict__ coef,
                 const float* __restrict__ bs,
                 float* __restrict__ out,
                 int mtiles, int ntiles, int nstations, int ntime,
                 int mAligned)
{
    int gwave = (int)((blockIdx.x * 256u + threadIdx.x) >> 5);
    int lane  = threadIdx.x & 31;
    if (gwave >= mtiles * ntiles) return;   // wave-uniform: EXEC stays all-1s

    int tm   = gwave / ntiles;
    int tn   = gwave - tm * ntiles;
    int half = lane >> 4;                   // 0: lanes 0-15, 1: lanes 16-31
    int l15  = lane & 15;

    // A: lane holds row (tm*16+l15), K = 4j + 2*half + {0,1}  -> one b64/step
    const float* aptr = coef + ((size_t)(tm * 16 + l15) * 16 + 2 * half);
    // B: lane holds col (tn*16+l15), same K pairing             -> one b64/step
    int col = tn * 16 + l15;
    const float* bptr = bs + ((size_t)half * BS_COLS + col) * 2;

    v8f acc = {};
    #pragma unroll
    for (int j = 0; j < 4; ++j) {
        v2f a = *(const v2f*)(aptr + 4 * j);
        v2f b = *(const v2f*)(bptr + (size_t)(2 * j) * BS_COLS * 2);
        acc = __builtin_amdgcn_wmma_f32_16x16x4_f32(
                  /*neg_a=*/false, a, /*neg_b=*/false, b,
                  /*c_mod=*/(short)0, acc, /*reuse_a=*/false, /*reuse_b=*/false);
    }

    // D layout: VGPR r -> M = tm*16 + half*8 + r ; N = l15 (col)
    if (col < ntime) {
        int    mbase = tm * 16 + half * 8;
        float* p     = out + (size_t)mbase * ntime + col;
        if (mAligned) {
            // nstations % 16 == 0: all 8 rows in range, flat NT-store run.
            #pragma unroll
            for (int r = 0; r < 8; ++r) {
                __builtin_nontemporal_store(acc[r], p + (size_t)r * ntime);
            }
        } else {
            #pragma unroll
            for (int r = 0; r < 8; ++r) {
                if (mbase + r < nstations) {
                    __builtin_nontemporal_store(acc[r], p + (size_t)r * ntime);
                }
            }
        }
    }
}

// ---------------------------------------------------------------------------
extern "C" void kernel_launch(void* const* d_in, const int* in_sizes, int n_in,
                              void* d_out, int out_size, void* d_ws, size_t ws_size,
                              hipStream_t stream)
{
    const float* tv     = (const float*)d_in[0];   // time_vector        [ntime]
    const float* c0     = (const float*)d_in[1];   // constant_offset    [ns]
    const float* trend  = (const float*)d_in[2];   // linear_trend       [ns]
    const float* samp   = (const float*)d_in[3];   // seasonal_amplitudes[ns,3]
    const float* sphase = (const float*)d_in[4];   // seasonal_phases    [ns,3]
    const float* ramp   = (const float*)d_in[5];   // residual_amplitudes[ns,2]
    const float* rphase = (const float*)d_in[6];   // residual_phases    [ns,2]
    const float* rper   = (const float*)d_in[7];   // residual_periods   [2]
    const float* per    = (const float*)d_in[8];   // periods            [3]
    const int*   nbidx  = (const int*)  d_in[9];   // neighbor_indices   [ns,8]
    const float* nbw    = (const float*)d_in[10];  // neighbor_weights   [ns,8]
    float*       out    = (float*)d_out;

    int ntime     = in_sizes[0];
    int nstations = in_sizes[1];
    int mtiles    = (nstations + 15) / 16;
    int ntiles    = (ntime + 15) / 16;
    int npad      = mtiles * 16;
    int mAligned  = (nstations % 16 == 0) ? 1 : 0;

    // Workspace: coef [npad x 16] f32, then basis [8 pairs x BS_COLS x 2] f32.
    float* coef = (float*)d_ws;
    size_t coefBytes = (((size_t)npad * 16 * sizeof(float)) + 255) & ~(size_t)255;
    float* bs   = (float*)((char*)d_ws + coefBytes);

    coef_kernel<<<(npad + 255) / 256, 256, 0, stream>>>(
        c0, trend, samp, sphase, ramp, rphase, nbidx, nbw, coef, nstations, npad);

    basis_kernel<<<BS_COLS / 256, 256, 0, stream>>>(tv, per, rper, bs, ntime);

    int tiles  = mtiles * ntiles;
    int blocks = (tiles + 7) / 8;          // 8 waves (tiles) per 256-thread block
    wmma_gemm_kernel<<<blocks, 256, 0, stream>>>(
        coef, bs, out, mtiles, ntiles, nstations, ntime, mAligned);
}